// Baseline_58205396795680
// MI455X (gfx1250) — compile-verified
//
#include <hip/hip_runtime.h>
#include <hip/hip_bf16.h>

// Problem constants from the reference
#define VR       4      // VOXEL_RES
#define FEAT     64     // VR^3
#define BATCH    64
#define NCLASS   40

typedef __attribute__((ext_vector_type(2))) float v2f;
typedef __attribute__((ext_vector_type(8))) float v8f;

// ---- monotonic f32 <-> u32 order mapping (for integer atomic min/max) ----
__device__ __forceinline__ unsigned f2ord(float f) {
  unsigned u = __float_as_uint(f);
  return (u & 0x80000000u) ? ~u : (u | 0x80000000u);
}
__device__ __forceinline__ float ord2f(unsigned u) {
  return __uint_as_float((u & 0x80000000u) ? (u ^ 0x80000000u) : ~u);
}

// ws layout (uint32): [0,192) mins  [192,384) maxs  [384, 384+BATCH*FEAT) counts
__global__ void init_ws_kernel(unsigned* __restrict__ ws) {
  int i = blockIdx.x * blockDim.x + threadIdx.x;
  const int nmm = 2 * BATCH * 3;                 // 384
  const int tot = nmm + BATCH * FEAT;            // 4480
  if (i < BATCH * 3)      ws[i] = 0xFFFFFFFFu;   // mins -> +max in ord space
  else if (i < nmm)       ws[i] = 0u;            // maxs -> -max in ord space
  else if (i < tot)       ws[i] = 0u;            // counts
}

// Pass 1: per-batch min/max over N points (grid: x=chunks, y=batch)
__global__ void minmax_kernel(const float* __restrict__ x,
                              unsigned* __restrict__ mins,
                              unsigned* __restrict__ maxs, int N) {
  const int b   = blockIdx.y;
  const int tid = threadIdx.x;
  const float* xb = x + (size_t)b * N * 3;
  float mn0 =  INFINITY, mn1 =  INFINITY, mn2 =  INFINITY;
  float mx0 = -INFINITY, mx1 = -INFINITY, mx2 = -INFINITY;
  const int stride = gridDim.x * blockDim.x;
  for (int n = blockIdx.x * blockDim.x + tid; n < N; n += stride) {
    float v0 = xb[n * 3 + 0], v1 = xb[n * 3 + 1], v2 = xb[n * 3 + 2];
    mn0 = fminf(mn0, v0); mx0 = fmaxf(mx0, v0);
    mn1 = fminf(mn1, v1); mx1 = fmaxf(mx1, v1);
    mn2 = fminf(mn2, v2); mx2 = fmaxf(mx2, v2);
  }
  __shared__ unsigned smin[3], smax[3];
  if (tid < 3) { smin[tid] = 0xFFFFFFFFu; smax[tid] = 0u; }
  __syncthreads();
  atomicMin(&smin[0], f2ord(mn0)); atomicMax(&smax[0], f2ord(mx0));
  atomicMin(&smin[1], f2ord(mn1)); atomicMax(&smax[1], f2ord(mx1));
  atomicMin(&smin[2], f2ord(mn2)); atomicMax(&smax[2], f2ord(mx2));
  __syncthreads();
  if (tid < 3) {
    atomicMin(&mins[b * 3 + tid], smin[tid]);
    atomicMax(&maxs[b * 3 + tid], smax[tid]);
  }
}

__device__ __forceinline__ int binidx(float v, float mn, float sc) {
  int i = (int)floorf((v - mn) * sc);
  i = i < 0 ? 0 : i;
  return i > (VR - 1) ? (VR - 1) : i;
}

// Pass 2: LDS-privatized 64-bin histogram per batch (x should be L2-resident)
__global__ void hist_kernel(const float* __restrict__ x,
                            const unsigned* __restrict__ mins,
                            const unsigned* __restrict__ maxs,
                            unsigned* __restrict__ counts, int N) {
  const int b   = blockIdx.y;
  const int tid = threadIdx.x;
  __shared__ unsigned h[FEAT];
  if (tid < FEAT) h[tid] = 0u;
  float mn[3], sc[3];
#pragma unroll
  for (int d = 0; d < 3; ++d) {
    float lo = ord2f(mins[b * 3 + d]);
    float hi = ord2f(maxs[b * 3 + d]);
    mn[d] = lo;
    float range = hi - lo;
    sc[d] = (range > 0.f) ? ((float)VR / range) : 0.f;
  }
  __syncthreads();
  const float* xb = x + (size_t)b * N * 3;
  const int stride = gridDim.x * blockDim.x;
  for (int n = blockIdx.x * blockDim.x + tid; n < N; n += stride) {
    int i0 = binidx(xb[n * 3 + 0], mn[0], sc[0]);
    int i1 = binidx(xb[n * 3 + 1], mn[1], sc[1]);
    int i2 = binidx(xb[n * 3 + 2], mn[2], sc[2]);
    atomicAdd(&h[i0 * 16 + i1 * 4 + i2], 1u);
  }
  __syncthreads();
  if (tid < FEAT) atomicAdd(&counts[b * FEAT + tid], h[tid]);
}

// Pass 3: out(64x40) = (counts/N)(64x64) @ W^T(64x40) + b, via V_WMMA_F32_16X16X4_F32.
// 12 waves: mt in [0,4) x nt in [0,3); N tiles padded to 48, lanes masked.
__global__ void gemm_wmma_kernel(const unsigned* __restrict__ counts,
                                 const float* __restrict__ W,
                                 const float* __restrict__ bias,
                                 float* __restrict__ out, float invN) {
  const int tid  = threadIdx.x;
  const int wave = tid >> 5;           // 0..11
  const int lane = tid & 31;
  const int mt   = wave / 3;           // 0..3  (row tile of batches)
  const int nt   = wave % 3;           // 0..2  (col tile of classes)
  const int half = lane >> 4;          // 0 or 1
  const int l16  = lane & 15;

  const int m  = mt * 16 + l16;        // A row (batch), always valid (<64)
  const int c  = nt * 16 + l16;        // B row (class), may be >= 40
  const int cc = (c < NCLASS) ? c : 0;
  const float bm = (c < NCLASS) ? 1.0f : 0.0f;

  v8f acc = {};
#pragma unroll
  for (int k0 = 0; k0 < FEAT; k0 += 4) {
    const int ka = k0 + 2 * half;      // lanes 0-15: K=k0,k0+1 ; lanes 16-31: K=k0+2,k0+3
    v2f a  = { (float)counts[m  * FEAT + ka] * invN,
               (float)counts[m  * FEAT + ka + 1] * invN };
    v2f bf = { W[cc * FEAT + ka] * bm,
               W[cc * FEAT + ka + 1] * bm };
    acc = __builtin_amdgcn_wmma_f32_16x16x4_f32(
        /*neg_a=*/false, a, /*neg_b=*/false, bf,
        /*c_mod=*/(short)0, acc, /*reuse_a=*/false, /*reuse_b=*/false);
  }

  const int n = nt * 16 + l16;         // D column (class)
  if (n < NCLASS) {
    float bv = bias[n];
#pragma unroll
    for (int v = 0; v < 8; ++v) {
      int mm = mt * 16 + v + 8 * half; // D row per ISA C/D layout
      out[mm * NCLASS + n] = acc[v] + bv;
    }
  }
}

extern "C" void kernel_launch(void* const* d_in, const int* in_sizes, int n_in,
                              void* d_out, int out_size, void* d_ws, size_t ws_size,
                              hipStream_t stream) {
  const float* x    = (const float*)d_in[0];   // (64, N, 3) f32
  const float* W    = (const float*)d_in[1];   // (40, 64)  f32
  const float* bias = (const float*)d_in[2];   // (40,)     f32
  float* out        = (float*)d_out;           // (64, 40)  f32

  const int N = in_sizes[0] / (BATCH * 3);

  unsigned* ws     = (unsigned*)d_ws;
  unsigned* mins   = ws;                       // 192
  unsigned* maxs   = ws + BATCH * 3;           // 192
  unsigned* counts = ws + 2 * BATCH * 3;       // 4096

  const int init_total = 2 * BATCH * 3 + BATCH * FEAT;   // 4480
  init_ws_kernel<<<(init_total + 255) / 256, 256, 0, stream>>>(ws);

  dim3 grid(64, BATCH);                        // 4096 blocks x 256 threads per pass
  minmax_kernel<<<grid, 256, 0, stream>>>(x, mins, maxs, N);
  hist_kernel  <<<grid, 256, 0, stream>>>(x, mins, maxs, counts, N);
  gemm_wmma_kernel<<<1, 384, 0, stream>>>(counts, W, bias, out, 1.0f / (float)N);
}